// SheafHyperGNN_89283780149647
// MI455X (gfx1250) — compile-verified
//
#include <hip/hip_runtime.h>
#include <math.h>

// ---------------------------------------------------------------------------
// SheafHyperGNN forward for MI455X (gfx1250), wave32, WMMA f32.
//
// Dims: N=20000, E=5000, NNZ=200000, d=6, HID=9, F_IN=256, OUT=4968, 2 layers.
//
// All GEMM operands are zero-padded in workspace so the WMMA K-loop has no
// masking / divergence. Each wave computes a 16x64 strip (4 accumulators
// sharing one A fragment), with double-buffered K-step fragments so loads of
// step k+1 overlap the 4 WMMAs of step k (no s_wait_loadcnt 0 per WMMA).
// ---------------------------------------------------------------------------

#define N_NODES 20000
#define N_EDGES 5000
#define E_PADR  5008             // 313*16 padded edge rows
#define NNZ     200000
#define DH      6
#define HID     9
#define F_IN    256
#define OUT_F   4968
#define PROJ    (DH * HID)       // 54
#define KPAD    64               // padded K / row stride of node state
#define NGROUPS 78               // ceil(311/4) groups of 4 N-tiles
#define WPAD_N  (NGROUPS * 64)   // 4992, padded lin2 width

typedef float v2f __attribute__((ext_vector_type(2)));
typedef float v8f __attribute__((ext_vector_type(8)));

// ---------------------------------------------------------------------------
// fp32 WMMA GEMM, one wave per 16x(16*NT) output strip.
//   C[M, Nn] = A[M, K] * B[K, *] + bias   (A row stride = K, compile-time)
// Guarantees (from padded workspace buffers):
//   - M multiple of 16, all A/B loads in bounds
//   - only the store is masked with nc < Nn
// ---------------------------------------------------------------------------
template <int K, int NT>
__global__ __launch_bounds__(32)
void wmma_gemm_f32(const float* __restrict__ A, const float* __restrict__ B,
                   const float* __restrict__ bias, float* __restrict__ C,
                   int Nn, int ldb, int ldc, int ngroups)
{
    const int g    = blockIdx.x;
    const int tm   = g / ngroups;
    const int tg   = g - tm * ngroups;
    const int lane = threadIdx.x;

    const int M0    = tm * 16;
    const int N0    = tg * (16 * NT);
    const int nlane = lane & 15;
    const int khalf = (lane >> 4) * 2;           // 0 or 2
    const int mrow  = M0 + nlane;

    const float* __restrict__ Arow = A + (size_t)mrow * K + khalf;
    const float* __restrict__ Bcol = B + N0 + nlane;

    v8f acc[NT];
#pragma unroll
    for (int t = 0; t < NT; ++t) acc[t] = (v8f){};

    auto loadA = [&](int k) -> v2f {
        // 2 contiguous floats per lane -> single b64 load (8B aligned)
        return *(const v2f*)(Arow + k);
    };
    auto loadB = [&](int k, v2f* b) {
        const float* __restrict__ p0 = Bcol + (size_t)(k + khalf) * ldb;
        const float* __restrict__ p1 = p0 + ldb;
#pragma unroll
        for (int t = 0; t < NT; ++t) { b[t].x = p0[t * 16]; b[t].y = p1[t * 16]; }
    };

    // double-buffered K-steps: prefetch k+4 while WMMAs consume k
    v2f aC = loadA(0), aN;
    v2f bC[NT], bN[NT];
    loadB(0, bC);

#pragma unroll 2
    for (int k0 = 0; k0 < K - 4; k0 += 4) {
        aN = loadA(k0 + 4);
        loadB(k0 + 4, bN);
#pragma unroll
        for (int t = 0; t < NT; ++t)
            acc[t] = __builtin_amdgcn_wmma_f32_16x16x4_f32(
                false, aC, false, bC[t], (short)0, acc[t], false, false);
        aC = aN;
#pragma unroll
        for (int t = 0; t < NT; ++t) bC[t] = bN[t];
    }
    // peeled last step (no prefetch)
#pragma unroll
    for (int t = 0; t < NT; ++t)
        acc[t] = __builtin_amdgcn_wmma_f32_16x16x4_f32(
            false, aC, false, bC[t], (short)0, acc[t], false, false);

    // C/D layout: VGPR i -> M = M0+i (lanes 0-15) / M0+8+i (lanes 16-31),
    // N = N0 + t*16 + (lane & 15)
    const int mbase = M0 + ((lane >> 4) ? 8 : 0);
#pragma unroll
    for (int t = 0; t < NT; ++t) {
        const int nc = N0 + t * 16 + nlane;
        if (nc < Nn) {
            const float bv = bias[nc];
#pragma unroll
            for (int i = 0; i < 8; ++i)
                C[(size_t)(mbase + i) * ldc + nc] = acc[t][i] + bv;
        }
    }
}

// he_attr [5000,256] -> [5008,256] zero-padded rows
__global__ void pad_edge_rows_kernel(const float* __restrict__ src,
                                     float* __restrict__ dst)
{
    int t = blockIdx.x * blockDim.x + threadIdx.x;
    if (t >= E_PADR * F_IN) return;
    int r = t / F_IN;
    dst[t] = (r < N_EDGES) ? src[t] : 0.0f;
}

// lin_w [256,54] -> [256,64] zero-padded cols
__global__ void pad_linw_kernel(const float* __restrict__ w,
                                float* __restrict__ wpad)
{
    int t = blockIdx.x * blockDim.x + threadIdx.x;
    if (t >= F_IN * KPAD) return;
    int k = t / KPAD, j = t - k * KPAD;
    wpad[t] = (j < PROJ) ? w[(size_t)k * PROJ + j] : 0.0f;
}

// lin2_w [54,4968] -> [64, WPAD_N] zero-padded
__global__ void pad_lin2w_kernel(const float* __restrict__ w,
                                 float* __restrict__ wpad)
{
    int t = blockIdx.x * blockDim.x + threadIdx.x;
    if (t >= KPAD * WPAD_N) return;
    int k = t / WPAD_N, j = t - k * WPAD_N;
    wpad[t] = (k < PROJ && j < OUT_F) ? w[(size_t)k * OUT_F + j] : 0.0f;
}

// Mean over the d=6 blocks of a 54-wide projection row (stride KPAD layout).
__global__ void mean_over_d_kernel(const float* __restrict__ proj,
                                   float* __restrict__ outm, int rows)
{
    int t = blockIdx.x * blockDim.x + threadIdx.x;
    if (t >= rows * HID) return;
    int r = t / HID, h = t - r * HID;
    const float* p = proj + (size_t)r * KPAD;
    float s = 0.0f;
#pragma unroll
    for (int dd = 0; dd < DH; ++dd) s += p[dd * HID + h];
    outm[(size_t)r * HID + h] = s * (1.0f / 6.0f);
}

// Incidence counts per node / per edge.
__global__ void degree_count_kernel(const int* __restrict__ node_idx,
                                    const int* __restrict__ edge_idx,
                                    float* __restrict__ ncnt,
                                    float* __restrict__ ecnt, int nnz)
{
    int i = blockIdx.x * blockDim.x + threadIdx.x;
    if (i >= nnz) return;
    atomicAdd(&ncnt[node_idx[i]], 1.0f);
    atomicAdd(&ecnt[edge_idx[i]], 1.0f);
}

// Per-incidence sheaf weights: sigmoid(LN(concat(xm,em)) @ W + b) -> alpha[i,0..5]
__global__ void sheaf_alpha_kernel(const float* __restrict__ xm,
                                   const float* __restrict__ em,
                                   const int* __restrict__ node_idx,
                                   const int* __restrict__ edge_idx,
                                   const float* __restrict__ g,
                                   const float* __restrict__ beta,
                                   const float* __restrict__ W,   // [18,6]
                                   const float* __restrict__ bb,  // [6]
                                   float* __restrict__ alpha, int nnz)
{
    int i = blockIdx.x * blockDim.x + threadIdx.x;
    if (i >= nnz) return;
    float v[2 * HID];
    const float* xp = xm + (size_t)node_idx[i] * HID;
    const float* ep = em + (size_t)edge_idx[i] * HID;
#pragma unroll
    for (int h = 0; h < HID; ++h) { v[h] = xp[h]; v[HID + h] = ep[h]; }

    float m = 0.0f;
#pragma unroll
    for (int j = 0; j < 2 * HID; ++j) m += v[j];
    m *= (1.0f / (2 * HID));
    float var = 0.0f;
#pragma unroll
    for (int j = 0; j < 2 * HID; ++j) { float dvv = v[j] - m; var += dvv * dvv; }
    var *= (1.0f / (2 * HID));
    const float rs = rsqrtf(var + 1e-5f);
#pragma unroll
    for (int j = 0; j < 2 * HID; ++j) v[j] = (v[j] - m) * rs * g[j] + beta[j];

#pragma unroll
    for (int c = 0; c < DH; ++c) {
        float s = bb[c];
#pragma unroll
        for (int j = 0; j < 2 * HID; ++j) s += v[j] * W[j * DH + c];
        alpha[(size_t)i * DH + c] = 1.0f / (1.0f + expf(-s));
    }
}

// h0 = LayerNorm(x_row) @ conv_w + conv_b, per expanded row r in [0, N*d).
__global__ void conv_h0_kernel(const float* __restrict__ x,   // [N, KPAD]
                               const float* __restrict__ g,
                               const float* __restrict__ beta,
                               const float* __restrict__ W,    // [9,9]
                               const float* __restrict__ bb,   // [9]
                               float* __restrict__ h0, int rows)
{
    int r = blockIdx.x * blockDim.x + threadIdx.x;
    if (r >= rows) return;
    const int n = r / DH, dd = r - n * DH;
    const float* xp = x + (size_t)n * KPAD + dd * HID;
    float v[HID];
#pragma unroll
    for (int h = 0; h < HID; ++h) v[h] = xp[h];

    float m = 0.0f;
#pragma unroll
    for (int h = 0; h < HID; ++h) m += v[h];
    m *= (1.0f / HID);
    float var = 0.0f;
#pragma unroll
    for (int h = 0; h < HID; ++h) { float dvv = v[h] - m; var += dvv * dvv; }
    var *= (1.0f / HID);
    const float rs = rsqrtf(var + 1e-5f);
#pragma unroll
    for (int h = 0; h < HID; ++h) v[h] = (v[h] - m) * rs * g[h] + beta[h];

    float* op = h0 + (size_t)r * HID;
#pragma unroll
    for (int c = 0; c < HID; ++c) {
        float s = bb[c];
#pragma unroll
        for (int j = 0; j < HID; ++j) s += v[j] * W[j * HID + c];
        op[c] = s;
    }
}

// node -> edge: m[edge*d+dr] += alpha * h0[node*d+dr]
__global__ void scatter_node_edge_kernel(const float* __restrict__ alpha,
                                         const int* __restrict__ node_idx,
                                         const int* __restrict__ edge_idx,
                                         const float* __restrict__ h0,
                                         float* __restrict__ macc, int nnz)
{
    int t = blockIdx.x * blockDim.x + threadIdx.x;
    if (t >= nnz * DH) return;
    int i = t / DH, dr = t - i * DH;
    const float a = alpha[(size_t)i * DH + dr];
    const size_t row = (size_t)node_idx[i] * DH + dr;
    const size_t col = (size_t)edge_idx[i] * DH + dr;
    const float* hp = h0 + row * HID;
    float* mp = macc + col * HID;
#pragma unroll
    for (int h = 0; h < HID; ++h) atomicAdd(&mp[h], a * hp[h]);
}

// scale expanded-edge rows by Be = 1/count (0 if empty)
__global__ void scale_edge_kernel(float* __restrict__ macc,
                                  const float* __restrict__ ecnt, int rows)
{
    int t = blockIdx.x * blockDim.x + threadIdx.x;
    if (t >= rows * HID) return;
    int r = t / HID;
    const float c = ecnt[r / DH];
    macc[t] *= (c > 0.0f) ? (1.0f / c) : 0.0f;
}

// edge -> node: acc[node*d+dr] += alpha * m[edge*d+dr]
__global__ void gather_edge_node_kernel(const float* __restrict__ alpha,
                                        const int* __restrict__ node_idx,
                                        const int* __restrict__ edge_idx,
                                        const float* __restrict__ macc,
                                        float* __restrict__ nacc, int nnz)
{
    int t = blockIdx.x * blockDim.x + threadIdx.x;
    if (t >= nnz * DH) return;
    int i = t / DH, dr = t - i * DH;
    const float a = alpha[(size_t)i * DH + dr];
    const size_t row = (size_t)node_idx[i] * DH + dr;
    const size_t col = (size_t)edge_idx[i] * DH + dr;
    const float* mp = macc + col * HID;
    float* op = nacc + row * HID;
#pragma unroll
    for (int h = 0; h < HID; ++h) atomicAdd(&op[h], a * mp[h]);
}

// x = [acc*Dv + h0 + conv_bias], optional ELU, written back to padded layout.
__global__ void conv_finalize_kernel(const float* __restrict__ nacc,
                                     const float* __restrict__ h0,
                                     const float* __restrict__ ncnt,
                                     const float* __restrict__ cbias,  // [9]
                                     float* __restrict__ x,            // [N,KPAD]
                                     int rows, int apply_elu)
{
    int t = blockIdx.x * blockDim.x + threadIdx.x;
    if (t >= rows * HID) return;
    int r = t / HID, h = t - r * HID;
    const int n = r / DH, dd = r - n * DH;
    const float c = ncnt[n];
    const float dv = (c > 0.0f) ? (1.0f / c) : 0.0f;
    float o = nacc[t] * dv + h0[t] + cbias[h];
    if (apply_elu) o = (o > 0.0f) ? o : expm1f(o);
    x[(size_t)n * KPAD + dd * HID + h] = o;
}

// ---------------------------------------------------------------------------
extern "C" void kernel_launch(void* const* d_in, const int* in_sizes, int n_in,
                              void* d_out, int out_size, void* d_ws, size_t ws_size,
                              hipStream_t stream)
{
    (void)in_sizes; (void)n_in; (void)out_size; (void)ws_size;

    const float* x_in      = (const float*)d_in[0];
    const float* he_attr   = (const float*)d_in[1];
    const int*   node_idx  = (const int*)d_in[2];
    const int*   edge_idx  = (const int*)d_in[3];
    const float* lin_w     = (const float*)d_in[4];
    const float* lin_b     = (const float*)d_in[5];
    const float* sheaf_g   = (const float*)d_in[6];
    const float* sheaf_bt  = (const float*)d_in[7];
    const float* sheaf_w   = (const float*)d_in[8];
    const float* sheaf_b   = (const float*)d_in[9];
    const float* conv_ln_g = (const float*)d_in[10];
    const float* conv_ln_b = (const float*)d_in[11];
    const float* conv_w    = (const float*)d_in[12];
    const float* conv_b    = (const float*)d_in[13];
    const float* conv_bias = (const float*)d_in[14];
    const float* lin2_w    = (const float*)d_in[15];
    const float* lin2_b    = (const float*)d_in[16];
    float*       out       = (float*)d_out;

    // ---- workspace partition (256B aligned slabs) ----
    size_t off = 0;
    auto alloc_f = [&](size_t nfl) -> float* {
        float* p = (float*)((char*)d_ws + off);
        off += ((nfl * sizeof(float) + 255) / 256) * 256;
        return p;
    };
    float* xpad    = alloc_f((size_t)N_NODES * KPAD);   // node state, padded
    float* eproj   = alloc_f((size_t)E_PADR * KPAD);    // edge projection, padded
    float* hepad   = alloc_f((size_t)E_PADR * F_IN);    // padded edge attrs
    float* linwpad = alloc_f((size_t)F_IN * KPAD);      // padded lin weights
    float* xm      = alloc_f((size_t)N_NODES * HID);
    float* em      = alloc_f((size_t)N_EDGES * HID);
    float* alpha   = alloc_f((size_t)NNZ * DH);
    float* ncnt    = alloc_f(N_NODES);
    float* ecnt    = alloc_f(N_EDGES);
    float* h0      = alloc_f((size_t)N_NODES * DH * HID);
    float* macc    = alloc_f((size_t)N_EDGES * DH * HID);
    float* nacc    = alloc_f((size_t)N_NODES * DH * HID);
    float* wpad    = alloc_f((size_t)KPAD * WPAD_N);    // padded lin2 weights

    const int TB = 256;
    auto blks = [](long long n, int tb) { return (unsigned)((n + tb - 1) / tb); };

    // zero pad region of node state (cols 54..63 must be 0 for final GEMM)
    hipMemsetAsync(xpad, 0, (size_t)N_NODES * KPAD * sizeof(float), stream);

    // ---- pad operands so WMMA loops need no masks ----
    pad_edge_rows_kernel<<<blks((long long)E_PADR * F_IN, TB), TB, 0, stream>>>(he_attr, hepad);
    pad_linw_kernel<<<blks((long long)F_IN * KPAD, TB), TB, 0, stream>>>(lin_w, linwpad);
    pad_lin2w_kernel<<<blks((long long)KPAD * WPAD_N, TB), TB, 0, stream>>>(lin2_w, wpad);

    // ---- input projections via fp32 WMMA (one wave = 16 rows x 64 cols) ----
    wmma_gemm_f32<F_IN, 4><<<N_NODES / 16, 32, 0, stream>>>(
        x_in, linwpad, lin_b, xpad, PROJ, KPAD, KPAD, /*ngroups=*/1);
    wmma_gemm_f32<F_IN, 4><<<E_PADR / 16, 32, 0, stream>>>(
        hepad, linwpad, lin_b, eproj, PROJ, KPAD, KPAD, /*ngroups=*/1);

    // ---- means over sheaf dimension ----
    mean_over_d_kernel<<<blks((long long)N_NODES * HID, TB), TB, 0, stream>>>(xpad, xm, N_NODES);
    mean_over_d_kernel<<<blks((long long)N_EDGES * HID, TB), TB, 0, stream>>>(eproj, em, N_EDGES);

    // ---- degree counts ----
    hipMemsetAsync(ncnt, 0, N_NODES * sizeof(float), stream);
    hipMemsetAsync(ecnt, 0, N_EDGES * sizeof(float), stream);
    degree_count_kernel<<<blks(NNZ, TB), TB, 0, stream>>>(node_idx, edge_idx, ncnt, ecnt, NNZ);

    // ---- sheaf sigmoid weights ----
    sheaf_alpha_kernel<<<blks(NNZ, TB), TB, 0, stream>>>(
        xm, em, node_idx, edge_idx, sheaf_g, sheaf_bt, sheaf_w, sheaf_b, alpha, NNZ);

    // ---- 2 diffusion conv layers ----
    const int rowsN = N_NODES * DH;   // 120000
    const int rowsE = N_EDGES * DH;   // 30000
    for (int l = 0; l < 2; ++l) {
        conv_h0_kernel<<<blks(rowsN, TB), TB, 0, stream>>>(
            xpad, conv_ln_g + l * HID, conv_ln_b + l * HID,
            conv_w + l * HID * HID, conv_b + l * HID, h0, rowsN);

        hipMemsetAsync(macc, 0, (size_t)rowsE * HID * sizeof(float), stream);
        scatter_node_edge_kernel<<<blks((long long)NNZ * DH, TB), TB, 0, stream>>>(
            alpha, node_idx, edge_idx, h0, macc, NNZ);
        scale_edge_kernel<<<blks((long long)rowsE * HID, TB), TB, 0, stream>>>(macc, ecnt, rowsE);

        hipMemsetAsync(nacc, 0, (size_t)rowsN * HID * sizeof(float), stream);
        gather_edge_node_kernel<<<blks((long long)NNZ * DH, TB), TB, 0, stream>>>(
            alpha, node_idx, edge_idx, macc, nacc, NNZ);

        conv_finalize_kernel<<<blks((long long)rowsN * HID, TB), TB, 0, stream>>>(
            nacc, h0, ncnt, conv_bias + l * HID, xpad, rowsN, /*apply_elu=*/(l == 0) ? 1 : 0);
    }

    // ---- final GEMM [20000, 64] x [64, 4992] -> [20000, 4968] + bias ----
    // grid: 1250 row tiles x 78 groups of 4 N-tiles; store masked at 4968.
    wmma_gemm_f32<KPAD, 4><<<(N_NODES / 16) * NGROUPS, 32, 0, stream>>>(
        xpad, wpad, lin2_b, out, OUT_F, WPAD_N, OUT_F, NGROUPS);
}